// SingleAnchorInfoNCE_75943611728237
// MI455X (gfx1250) — compile-verified
//
#include <hip/hip_runtime.h>
#include <stdint.h>

typedef __attribute__((ext_vector_type(2))) float v2f;
typedef __attribute__((ext_vector_type(8))) float v8f;

#define BN 8192
#define DN 128
#define NCHUNK 128
#define PITCH 132   // floats per LDS row: stride 132 % 64 banks = 4 -> conflict-free b64 reads
#define NCHUNKS (BN / NCHUNK)

// ---------------------------------------------------------------------------
// Kernel 1: squared row norms of sk (rows 0..8191) and im (rows 8192..16383).
// One wave per row, 4 floats per lane, wave32 shuffle reduction.
// ---------------------------------------------------------------------------
__global__ __launch_bounds__(256) void norms_kernel(const float* __restrict__ sk,
                                                    const float* __restrict__ im,
                                                    float* __restrict__ norms) {
  int row  = blockIdx.x * 8 + (threadIdx.x >> 5);
  int lane = threadIdx.x & 31;
  const float* src = (row < BN) ? (sk + (size_t)row * DN)
                                : (im + (size_t)(row - BN) * DN);
  float4 v = *(const float4*)(src + lane * 4);
  float s = v.x * v.x + v.y * v.y + v.z * v.z + v.w * v.w;
#pragma unroll
  for (int m = 1; m < 32; m <<= 1) s += __shfl_xor(s, m, 32);
  if (lane == 0) norms[row] = s;
}

// ---------------------------------------------------------------------------
// Async stage of one 128x128 fp32 im chunk into an LDS buffer (padded rows)
// using the CDNA5 fire-and-forget global->LDS path (tracked by ASYNCcnt).
// 128 threads x 32 issues of B128: 64 KB per chunk, no VGPR round-trip.
// GVS addressing: SGPR pair = im base, per-lane 32-bit byte offset.
// ---------------------------------------------------------------------------
__device__ __forceinline__ void stage_async(float* __restrict__ buf,
                                            const float* __restrict__ im,
                                            int jbase, int tid) {
#pragma unroll
  for (int it = 0; it < 32; ++it) {
    int i   = tid + it * 128;
    int row = i >> 5;          // DN/4 = 32 float4 per row
    int c4  = i & 31;
    // generic shared pointer truncates to the 32-bit LDS byte address
    unsigned laddr = (unsigned)(uintptr_t)(buf + row * PITCH + c4 * 4);
    unsigned goff  = (unsigned)(((unsigned)(jbase + row) * DN + c4 * 4) * 4u);
    asm volatile("global_load_async_to_lds_b128 %0, %1, %2 offset:0"
                 :: "v"(laddr), "v"(goff), "s"(im)
                 : "memory");
  }
}

// ---------------------------------------------------------------------------
// Kernel 2: fused  -d(sk_i, im_j)  ->  sum_j exp(-d)  and diagonal d_ii.
// Each wave: 16 sk rows (A frag resident in 64 VGPRs), block of 4 waves: 64
// rows. im double-buffered in LDS via async copies; 32x chained
// V_WMMA_F32_16X16X4_F32 per 16x16 output tile (K = 128).
// ---------------------------------------------------------------------------
__global__ __launch_bounds__(128) void gemm_lse_kernel(const float* __restrict__ sk,
                                                       const float* __restrict__ im,
                                                       const float* __restrict__ norms,
                                                       float* __restrict__ partials) {
  __shared__ float lds[2][NCHUNK * PITCH];   // 2 x 67,584 B of 320 KB/WGP
  __shared__ float wsum[8];                  // 4 waves x 2 halves

  const int tid   = threadIdx.x;
  const int wave  = tid >> 5;
  const int lane  = tid & 31;
  const int half  = lane >> 4;     // 0: lanes 0-15, 1: lanes 16-31
  const int l15   = lane & 15;
  const int mbase = blockIdx.x * 64 + wave * 16;

  // A fragment: 16x128 fp32 strip, 32 k-steps of float2 per lane (64 VGPRs).
  // A(16x4) layout: lanes 0-15 hold K={4s,4s+1}, lanes 16-31 hold K={4s+2,4s+3}.
  v2f a[32];
  const float* skrow = sk + (size_t)(mbase + l15) * DN + 2 * half;
#pragma unroll
  for (int s = 0; s < 32; ++s) a[s] = *(const v2f*)(skrow + 4 * s);

  // C/D layout: VGPR r -> M = r + 8*half, N = l15.
  float skn[8];
#pragma unroll
  for (int r = 0; r < 8; ++r) skn[r] = norms[mbase + 8 * half + r];

  v8f acc_e = {0, 0, 0, 0, 0, 0, 0, 0};   // per-lane partial sum of exp(-d)
  v8f acc_d = {0, 0, 0, 0, 0, 0, 0, 0};   // diagonal distance (one lane per row)

  // Prefetch chunk 0.
  stage_async(&lds[0][0], im, 0, tid);

  for (int chunk = 0; chunk < NCHUNKS; ++chunk) {
    const int jbase = chunk * NCHUNK;
    float* cur = &lds[chunk & 1][0];

    // Drain this wave's async copies, then publish the buffer block-wide.
    asm volatile("s_wait_asynccnt 0x0" ::: "memory");
    __syncthreads();

    // Overlap: kick off the next chunk's copy while computing on this one.
    if (chunk + 1 < NCHUNKS)
      stage_async(&lds[(chunk + 1) & 1][0], im, jbase + NCHUNK, tid);

    for (int jt = 0; jt < NCHUNK; jt += 16) {
      float imn = norms[BN + jbase + jt + l15];
      // B(4x16) layout mirrors A: same float2 pattern, rows from im tile.
      const float* brow = &cur[(jt + l15) * PITCH + 2 * half];
      v8f c = {0, 0, 0, 0, 0, 0, 0, 0};
#pragma unroll
      for (int s = 0; s < 32; ++s) {
        v2f b = *(const v2f*)(brow + 4 * s);
        c = __builtin_amdgcn_wmma_f32_16x16x4_f32(
            /*neg_a=*/false, a[s], /*neg_b=*/false, b,
            /*c_mod=*/(short)0, c, /*reuse_a=*/false, /*reuse_b=*/false);
      }
      const int ng = jbase + jt + l15;
#pragma unroll
      for (int r = 0; r < 8; ++r) {
        float sq = skn[r] + imn - 2.0f * c[r];
        sq = fmaxf(sq, 0.0f);
        float dd = sqrtf(sq);
        acc_e[r] += __expf(-dd);                 // logits <= 0: no overflow risk
        const int mg = mbase + 8 * half + r;
        acc_d[r] = (ng == mg) ? dd : acc_d[r];   // v_cndmask, no EXEC change
      }
    }
    // Buffer reuse is safe: buf[chunk&1] is only re-issued after the barrier
    // at the top of iteration chunk+1, i.e. after all waves finished reading.
  }

  // Reduce across the 16 lanes of each half (masks < 16 never cross bit 4).
#pragma unroll
  for (int m = 1; m < 16; m <<= 1) {
#pragma unroll
    for (int r = 0; r < 8; ++r) {
      acc_e[r] += __shfl_xor(acc_e[r], m, 32);
      acc_d[r] += __shfl_xor(acc_d[r], m, 32);
    }
  }

  // loss_i = log(sum_j exp(-d_ij)) + d_ii ; each half-wave owns 8 rows.
  float wloss = 0.0f;
#pragma unroll
  for (int r = 0; r < 8; ++r) wloss += __logf(acc_e[r]) + acc_d[r];

  __syncthreads();
  if (l15 == 0) wsum[wave * 2 + half] = wloss;
  __syncthreads();
  if (tid == 0) {
    float t = 0.0f;
#pragma unroll
    for (int i = 0; i < 8; ++i) t += wsum[i];
    partials[blockIdx.x] = t;   // fixed-order: deterministic across replays
  }
}

// ---------------------------------------------------------------------------
// Kernel 3: deterministic final mean over the 128 block partials.
// ---------------------------------------------------------------------------
__global__ __launch_bounds__(128) void finalize_kernel(const float* __restrict__ partials,
                                                       float* __restrict__ out) {
  __shared__ float s[128];
  int t = threadIdx.x;
  s[t] = partials[t];
  __syncthreads();
  for (int k = 64; k > 0; k >>= 1) {
    if (t < k) s[t] += s[t + k];
    __syncthreads();
  }
  if (t == 0) out[0] = s[0] * (1.0f / 8192.0f);
}

extern "C" void kernel_launch(void* const* d_in, const int* in_sizes, int n_in,
                              void* d_out, int out_size, void* d_ws, size_t ws_size,
                              hipStream_t stream) {
  const float* sk = (const float*)d_in[0];
  const float* im = (const float*)d_in[1];
  float* norms    = (float*)d_ws;           // 16384 floats
  float* partials = norms + 2 * BN;         // 128 floats
  float* out      = (float*)d_out;

  norms_kernel<<<(2 * BN) / 8, 256, 0, stream>>>(sk, im, norms);
  gemm_lse_kernel<<<BN / 64, 128, 0, stream>>>(sk, im, norms, partials);
  finalize_kernel<<<1, 128, 0, stream>>>(partials, out);
}